// AE_88484916232447
// MI455X (gfx1250) — compile-verified
//
#include <hip/hip_runtime.h>
#include <math.h>

typedef __attribute__((ext_vector_type(2))) float v2f;
typedef __attribute__((ext_vector_type(8))) float v8f;

// ---------------------------------------------------------------------------
// Spiral convolution as WMMA f32 GEMM (branchless inner loop, M-blocked):
//   out[b,v,n] = sum_{s,c} h[b, idx[v,s], c] * W[n, s*C+c] + bias[n]
// One wave computes MT 16-row tiles x OC columns; K stepped by 4 with
// V_WMMA_F32_16X16X4_F32. W fragments are reused across the MT row tiles.
// C must be a multiple of 4 (pre-padded). Columns n >= OC load from a
// clamped W row; their accumulator lanes are garbage but never stored.
// ---------------------------------------------------------------------------
template <int C, int OC, int ELU, int MT>
__global__ __launch_bounds__(256) void spiral_conv_wmma(
    const float* __restrict__ h,    // (64, Nv, C)
    const int* __restrict__ idx,    // (Nv, 9)
    const float* __restrict__ W,    // (OC, 9*C)
    const float* __restrict__ bias, // (OC)
    float* __restrict__ out,        // (64, Nv, OC)
    int Nv, int superTiles) {
  constexpr int NCH = (OC + 15) / 16;
  constexpr int KROW = 9 * C;
  static_assert((C & 3) == 0, "C must be padded to a multiple of 4");

  const int lane = threadIdx.x & 31;
  const int wv = threadIdx.x >> 5;
  const int st = blockIdx.x * 8 + wv;
  if (st >= superTiles) return;  // whole-wave exit: EXEC stays all-ones

  const int l15 = lane & 15;
  const int hi = lane >> 4;   // 0 for lanes 0-15, 1 for 16-31
  const int koff = hi << 1;   // A/B fragment K offset {0,2}

  // Per row-tile state
  const int* __restrict__ vidx[MT];
  const float* __restrict__ hb[MT];
#pragma unroll
  for (int t = 0; t < MT; ++t) {
    const long row = (long)(st * MT + t) * 16 + l15;  // A row this lane feeds
    const int b = (int)(row / Nv);
    const int v = (int)(row - (long)b * Nv);
    vidx[t] = idx + v * 9;
    hb[t] = h + (long)b * Nv * C;
  }

  v8f acc[MT][NCH];
  const float* __restrict__ wrow[NCH];
#pragma unroll
  for (int j = 0; j < NCH; ++j) {
    const int n = j * 16 + l15;
    const int nw = (n < OC) ? n : (OC - 1);  // clamp (folds away if OC%16==0)
    const float bv = bias[nw];
    wrow[j] = W + (long)nw * KROW;
#pragma unroll
    for (int t = 0; t < MT; ++t)
      acc[t][j] = (v8f){bv, bv, bv, bv, bv, bv, bv, bv};
  }

  for (int s = 0; s < 9; ++s) {
    const float* __restrict__ hrow[MT];
#pragma unroll
    for (int t = 0; t < MT; ++t) hrow[t] = hb[t] + (long)vidx[t][s] * C;
    const int kbase = s * C;
#pragma unroll
    for (int c0 = 0; c0 < C; c0 += 4) {
      const int ca = c0 + koff;
      v2f a[MT];
#pragma unroll
      for (int t = 0; t < MT; ++t) {
        a[t].x = hrow[t][ca];
        a[t].y = hrow[t][ca + 1];
      }
#pragma unroll
      for (int j = 0; j < NCH; ++j) {
        v2f bf;
        bf.x = wrow[j][kbase + ca];
        bf.y = wrow[j][kbase + ca + 1];
#pragma unroll
        for (int t = 0; t < MT; ++t) {
          acc[t][j] = __builtin_amdgcn_wmma_f32_16x16x4_f32(
              false, a[t], false, bf, (short)0, acc[t][j], false, false);
        }
      }
    }
  }

#pragma unroll
  for (int t = 0; t < MT; ++t) {
    const long obase = (long)(st * MT + t) * 16;
#pragma unroll
    for (int j = 0; j < NCH; ++j) {
      const int n = j * 16 + l15;
      if (n < OC) {
#pragma unroll
        for (int r = 0; r < 8; ++r) {
          const int m = r + (hi << 3);  // C/D layout: lanes>=16 hold M=r+8
          float val = acc[t][j][r];
          if (ELU) val = (val > 0.0f) ? val : (expf(val) - 1.0f);
          out[(obase + m) * OC + n] = val;
        }
      }
    }
  }
}

// ---------------------------------------------------------------------------
// Dense GEMM (M=64 rows): out(M,N) = A(M,K) @ W(N,K)^T + bias, K % 4 == 0.
// Invalid columns (n >= N) read a clamped W row and are never stored.
// ---------------------------------------------------------------------------
__global__ __launch_bounds__(256) void dense_wmma(
    const float* __restrict__ A, const float* __restrict__ W,
    const float* __restrict__ bias, float* __restrict__ out,
    int K, int N, int ntiles, int tilesTotal) {
  const int lane = threadIdx.x & 31;
  const int wv = threadIdx.x >> 5;
  const int tile = blockIdx.x * 8 + wv;
  if (tile >= tilesTotal) return;
  const int mt = tile / ntiles;
  const int nt = tile - mt * ntiles;
  const int l15 = lane & 15;
  const int hi = lane >> 4;
  const int koff = hi << 1;

  const int arow = mt * 16 + l15;
  const int n = nt * 16 + l15;
  const int nw = (n < N) ? n : (N - 1);
  const float bv = bias[nw];
  v8f acc = {bv, bv, bv, bv, bv, bv, bv, bv};

  const float* __restrict__ ap = A + (long)arow * K + koff;
  const float* __restrict__ wp = W + (long)nw * K + koff;
#pragma unroll 4
  for (int k0 = 0; k0 < K; k0 += 4) {
    v2f a, bf;
    a.x = ap[k0];
    a.y = ap[k0 + 1];
    bf.x = wp[k0];
    bf.y = wp[k0 + 1];
    acc = __builtin_amdgcn_wmma_f32_16x16x4_f32(false, a, false, bf, (short)0,
                                                acc, false, false);
  }
  if (n < N) {
#pragma unroll
    for (int r = 0; r < 8; ++r) {
      const int m = r + (hi << 3);
      out[(long)(mt * 16 + m) * N + n] = acc[r];
    }
  }
}

// ---------------------------------------------------------------------------
// Channel padding helpers for the first conv (C=3 -> 4)
// ---------------------------------------------------------------------------
__global__ void pad_x_kernel(const float* __restrict__ x,
                             float* __restrict__ xp, int rows) {
  const int t = blockIdx.x * 256 + threadIdx.x;
  if (t >= rows) return;
  xp[t * 4 + 0] = x[t * 3 + 0];
  xp[t * 4 + 1] = x[t * 3 + 1];
  xp[t * 4 + 2] = x[t * 3 + 2];
  xp[t * 4 + 3] = 0.0f;
}

__global__ void pad_w0_kernel(const float* __restrict__ W,  // (32, 27)
                              float* __restrict__ Wp) {     // (32, 36)
  const int t = blockIdx.x * 256 + threadIdx.x;
  if (t >= 32 * 9) return;
  const int n = t / 9;
  const int s = t - n * 9;
  Wp[n * 36 + s * 4 + 0] = W[n * 27 + s * 3 + 0];
  Wp[n * 36 + s * 4 + 1] = W[n * 27 + s * 3 + 1];
  Wp[n * 36 + s * 4 + 2] = W[n * 27 + s * 3 + 2];
  Wp[n * 36 + s * 4 + 3] = 0.0f;
}

// ---------------------------------------------------------------------------
// Sparse pool: out[b, row[k], :] += val[k] * h[b, col[k], :]
// ---------------------------------------------------------------------------
__global__ void pool_scatter(const float* __restrict__ h,
                             const int* __restrict__ row,
                             const int* __restrict__ col,
                             const float* __restrict__ val,
                             float* __restrict__ out, int Nin, int Nout, int Ch,
                             int nnz) {
  const int t = blockIdx.x * 256 + threadIdx.x;
  const int total = nnz * 64;
  if (t >= total) return;
  const int k = t % nnz;
  const int b = t / nnz;
  const int r = row[k];
  const int c = col[k];
  const float vv = val[k];
  const float* __restrict__ src = h + ((long)b * Nin + c) * Ch;
  float* __restrict__ dst = out + ((long)b * Nout + r) * Ch;
  for (int ch = 0; ch < Ch; ++ch) {
    __hip_atomic_fetch_add(&dst[ch], vv * src[ch], __ATOMIC_RELAXED,
                           __HIP_MEMORY_SCOPE_AGENT);
  }
}

// ---------------------------------------------------------------------------
// JAX threefry2x32 dropout mask (key(1234), shape (64,1728)), gender_z = hf*mask
// ---------------------------------------------------------------------------
__global__ void genderz_kernel(const float* __restrict__ hf,
                               float* __restrict__ gz) {
  const int j = blockIdx.x * 256 + threadIdx.x;
  const int half = 55296;  // 110592 / 2
  if (j >= half) return;
  unsigned x0 = (unsigned)j;
  unsigned x1 = (unsigned)(j + half);
  const unsigned ks[3] = {0u, 1234u, 0u ^ 1234u ^ 0x1BD11BDAu};
  const int rot[2][4] = {{13, 15, 26, 6}, {17, 29, 16, 24}};
  x0 += ks[0];
  x1 += ks[1];
#pragma unroll
  for (int i = 0; i < 5; ++i) {
#pragma unroll
    for (int rr = 0; rr < 4; ++rr) {
      const int r = rot[i & 1][rr];
      x0 += x1;
      x1 = (x1 << r) | (x1 >> (32 - r));
      x1 ^= x0;
    }
    x0 += ks[(i + 1) % 3];
    x1 += ks[(i + 2) % 3] + (unsigned)(i + 1);
  }
  const float u0 = __uint_as_float((x0 >> 9) | 0x3f800000u) - 1.0f;
  const float u1 = __uint_as_float((x1 >> 9) | 0x3f800000u) - 1.0f;
  gz[j] = hf[j] * ((u0 >= 0.5f) ? 2.0f : 0.0f);
  gz[j + half] = hf[j + half] * ((u1 >= 0.5f) ? 2.0f : 0.0f);
}

// ---------------------------------------------------------------------------
// Head: out[b] = act(dot(gz[b,:1728], w) + b0), one wave per batch row
// ---------------------------------------------------------------------------
__global__ void head_dot(const float* __restrict__ gz,
                         const float* __restrict__ w,
                         const float* __restrict__ wb, float* __restrict__ out,
                         int sigmoid) {
  const int b = blockIdx.x;
  float s = 0.0f;
  for (int f = threadIdx.x; f < 1728; f += 32) s += gz[b * 1728 + f] * w[f];
  for (int o = 16; o > 0; o >>= 1) s += __shfl_down(s, o, 32);
  if (threadIdx.x == 0) {
    float r = s + wb[0];
    if (sigmoid) r = 1.0f / (1.0f + expf(-r));
    out[b] = r;
  }
}

// ---------------------------------------------------------------------------
// Latent generators: out[b,j] = src[b*stride+off] * W[j] + bias[j], (64,18)
// ---------------------------------------------------------------------------
__global__ void latgen(const float* __restrict__ src, int stride, int off,
                       const float* __restrict__ W,
                       const float* __restrict__ bias,
                       float* __restrict__ out) {
  const int t = blockIdx.x * 256 + threadIdx.x;
  if (t >= 64 * 18) return;
  const int b = t / 18;
  const int j = t - b * 18;
  out[t] = src[b * stride + off] * W[j] + bias[j];
}

static inline unsigned cdivu(long a, long b) { return (unsigned)((a + b - 1) / b); }

extern "C" void kernel_launch(void* const* d_in, const int* in_sizes, int n_in,
                              void* d_out, int out_size, void* d_ws,
                              size_t ws_size, hipStream_t stream) {
  (void)in_sizes; (void)n_in; (void)out_size; (void)ws_size;
  // ---- inputs (setup_inputs order, nested containers flattened) ----
  const float* x = (const float*)d_in[0];
  const float* y = (const float*)d_in[1];
  const float* pos = (const float*)d_in[2];
  const int* sp[4] = {(const int*)d_in[3], (const int*)d_in[4],
                      (const int*)d_in[5], (const int*)d_in[6]};
  const int *dR[4], *dC[4];
  const float* dV[4];
  for (int i = 0; i < 4; ++i) {
    dR[i] = (const int*)d_in[7 + 3 * i];
    dC[i] = (const int*)d_in[8 + 3 * i];
    dV[i] = (const float*)d_in[9 + 3 * i];
  }
  const int *uR[4], *uC[4];
  const float* uV[4];
  for (int i = 0; i < 4; ++i) {
    uR[i] = (const int*)d_in[19 + 3 * i];
    uC[i] = (const int*)d_in[20 + 3 * i];
    uV[i] = (const float*)d_in[21 + 3 * i];
  }
  const float* enW[4] = {(const float*)d_in[31], (const float*)d_in[32],
                         (const float*)d_in[33], (const float*)d_in[34]};
  const float* enB[4] = {(const float*)d_in[35], (const float*)d_in[36],
                         (const float*)d_in[37], (const float*)d_in[38]};
  const float* en_fc_W = (const float*)d_in[39];
  const float* en_fc_b = (const float*)d_in[40];
  const float* reb_W = (const float*)d_in[41];
  const float* reb_b = (const float*)d_in[42];
  const float* rew_W = (const float*)d_in[43];
  const float* rew_b = (const float*)d_in[44];
  const float* reh_W = (const float*)d_in[45];
  const float* reh_b = (const float*)d_in[46];
  const float* cl_W = (const float*)d_in[47];
  const float* cl_b = (const float*)d_in[48];
  const float* de_fc_W = (const float*)d_in[49];
  const float* de_fc_b = (const float*)d_in[50];
  const float* deW[4] = {(const float*)d_in[51], (const float*)d_in[52],
                         (const float*)d_in[53], (const float*)d_in[54]};
  const float* deB[4] = {(const float*)d_in[55], (const float*)d_in[56],
                         (const float*)d_in[57], (const float*)d_in[58]};
  const float* de_last_W = (const float*)d_in[59];
  const float* de_last_b = (const float*)d_in[60];
  const float* lgW[4] = {(const float*)d_in[61], (const float*)d_in[63],
                         (const float*)d_in[65], (const float*)d_in[67]};
  const float* lgB[4] = {(const float*)d_in[62], (const float*)d_in[64],
                         (const float*)d_in[66], (const float*)d_in[68]};

  // ---- output offsets (tuple order, flat) ----
  float* outp = (float*)d_out;
  const long OFF_OUT = 0;                       // (64,6890,3)
  const long OFF_GENDER = 1322880;              // (64,1)
  const long OFF_BZ = OFF_GENDER + 64;          // (64,18)
  const long OFF_GZL = OFF_BZ + 1152;
  const long OFF_WZ = OFF_GZL + 1152;
  const long OFF_HZ = OFF_WZ + 1152;
  const long OFF_Z = OFF_HZ + 1152;             // (64,72)
  const long OFF_GENDERZ = OFF_Z + 4608;        // (64,1728)
  const long OFF_BMI = OFF_GENDERZ + 110592;
  const long OFF_WEI = OFF_BMI + 64;
  const long OFF_HEI = OFF_WEI + 64;

  // ---- workspace ----
  const long BUFE = 64L * 6890 * 32;  // 14,110,720 floats per ping-pong buffer
  float* buf0 = (float*)d_ws;
  float* buf1 = buf0 + BUFE;
  float* hf = buf1 + BUFE;            // (64, 1728)
  float* xp = hf + 110592;            // (64, 6890, 4) channel-padded x
  float* w0p = xp + 64L * 6890 * 4;   // (32, 36) channel-padded en_W[0]

  const dim3 B256(256);

  // super-tile counts: (64*Nv/16)/2 — all exact since 64*Nv % 32 == 0
  const int ST0 = 13780;  // Nv=6890
  const int ST1 = 3446;   // Nv=1723
  const int ST2 = 862;    // Nv=431
  const int ST3 = 216;    // Nv=108

  // ===================== Encoder =====================
  pad_x_kernel<<<cdivu(440960, 256), B256, 0, stream>>>(x, xp, 440960);
  pad_w0_kernel<<<2, B256, 0, stream>>>(enW[0], w0p);
  // conv0: xp(64,6890,4) -> buf0(64,6890,32)
  spiral_conv_wmma<4, 32, 1, 2><<<cdivu(ST0, 8), B256, 0, stream>>>(
      xp, sp[0], w0p, enB[0], buf0, 6890, ST0);
  hipMemsetAsync(buf1, 0, 64L * 1723 * 32 * 4, stream);
  pool_scatter<<<cdivu(5169L * 64, 256), B256, 0, stream>>>(
      buf0, dR[0], dC[0], dV[0], buf1, 6890, 1723, 32, 5169);
  spiral_conv_wmma<32, 32, 1, 2><<<cdivu(ST1, 8), B256, 0, stream>>>(
      buf1, sp[1], enW[1], enB[1], buf0, 1723, ST1);
  hipMemsetAsync(buf1, 0, 64L * 431 * 32 * 4, stream);
  pool_scatter<<<cdivu(1293L * 64, 256), B256, 0, stream>>>(
      buf0, dR[1], dC[1], dV[1], buf1, 1723, 431, 32, 1293);
  spiral_conv_wmma<32, 32, 1, 2><<<cdivu(ST2, 8), B256, 0, stream>>>(
      buf1, sp[2], enW[2], enB[2], buf0, 431, ST2);
  hipMemsetAsync(buf1, 0, 64L * 108 * 32 * 4, stream);
  pool_scatter<<<cdivu(324L * 64, 256), B256, 0, stream>>>(
      buf0, dR[2], dC[2], dV[2], buf1, 431, 108, 32, 324);
  spiral_conv_wmma<32, 64, 1, 2><<<cdivu(ST3, 8), B256, 0, stream>>>(
      buf1, sp[3], enW[3], enB[3], buf0, 108, ST3);
  hipMemsetAsync(hf, 0, 110592L * 4, stream);
  pool_scatter<<<cdivu(81L * 64, 256), B256, 0, stream>>>(
      buf0, dR[3], dC[3], dV[3], hf, 108, 27, 64, 81);

  // z = hf @ en_fc_W^T + b : M=64 K=1728 N=72 (5 n-tiles, 20 tiles)
  dense_wmma<<<cdivu(20, 8), B256, 0, stream>>>(hf, en_fc_W, en_fc_b,
                                                outp + OFF_Z, 1728, 72, 5, 20);
  // dropout mask + gender_z
  genderz_kernel<<<cdivu(55296, 256), B256, 0, stream>>>(hf, outp + OFF_GENDERZ);
  // regression / classification heads
  head_dot<<<64, 32, 0, stream>>>(outp + OFF_GENDERZ, reb_W, reb_b, outp + OFF_BMI, 0);
  head_dot<<<64, 32, 0, stream>>>(outp + OFF_GENDERZ, rew_W, rew_b, outp + OFF_WEI, 0);
  head_dot<<<64, 32, 0, stream>>>(outp + OFF_GENDERZ, reh_W, reh_b, outp + OFF_HEI, 0);
  head_dot<<<64, 32, 0, stream>>>(outp + OFF_GENDERZ, cl_W, cl_b, outp + OFF_GENDER, 1);
  // latent generators
  latgen<<<cdivu(1152, 256), B256, 0, stream>>>(pos, 3, 0, lgW[0], lgB[0], outp + OFF_BZ);
  latgen<<<cdivu(1152, 256), B256, 0, stream>>>(y, 1, 0, lgW[1], lgB[1], outp + OFF_GZL);
  latgen<<<cdivu(1152, 256), B256, 0, stream>>>(pos, 3, 1, lgW[2], lgB[2], outp + OFF_WZ);
  latgen<<<cdivu(1152, 256), B256, 0, stream>>>(pos, 3, 2, lgW[3], lgB[3], outp + OFF_HZ);

  // ===================== Decoder =====================
  // d = z @ de_fc_W^T + b : M=64 K=72 N=1728 (108 n-tiles, 432 tiles) -> buf1
  dense_wmma<<<cdivu(432, 8), B256, 0, stream>>>(outp + OFF_Z, de_fc_W, de_fc_b,
                                                 buf1, 72, 1728, 108, 432);
  // i=0 (lev 3): pool ups[3] 27->108 ch64, conv (64->64)
  hipMemsetAsync(buf0, 0, 64L * 108 * 64 * 4, stream);
  pool_scatter<<<cdivu(324L * 64, 256), B256, 0, stream>>>(
      buf1, uR[3], uC[3], uV[3], buf0, 27, 108, 64, 324);
  spiral_conv_wmma<64, 64, 1, 2><<<cdivu(ST3, 8), B256, 0, stream>>>(
      buf0, sp[3], deW[0], deB[0], buf1, 108, ST3);
  // i=1 (lev 2): pool ups[2] 108->431 ch64, conv (64->32)
  hipMemsetAsync(buf0, 0, 64L * 431 * 64 * 4, stream);
  pool_scatter<<<cdivu(1293L * 64, 256), B256, 0, stream>>>(
      buf1, uR[2], uC[2], uV[2], buf0, 108, 431, 64, 1293);
  spiral_conv_wmma<64, 32, 1, 2><<<cdivu(ST2, 8), B256, 0, stream>>>(
      buf0, sp[2], deW[1], deB[1], buf1, 431, ST2);
  // i=2 (lev 1): pool ups[1] 431->1723 ch32, conv (32->32)
  hipMemsetAsync(buf0, 0, 64L * 1723 * 32 * 4, stream);
  pool_scatter<<<cdivu(5169L * 64, 256), B256, 0, stream>>>(
      buf1, uR[1], uC[1], uV[1], buf0, 431, 1723, 32, 5169);
  spiral_conv_wmma<32, 32, 1, 2><<<cdivu(ST1, 8), B256, 0, stream>>>(
      buf0, sp[1], deW[2], deB[2], buf1, 1723, ST1);
  // i=3 (lev 0): pool ups[0] 1723->6890 ch32, conv (32->32)
  hipMemsetAsync(buf0, 0, 64L * 6890 * 32 * 4, stream);
  pool_scatter<<<cdivu(20670L * 64, 256), B256, 0, stream>>>(
      buf1, uR[0], uC[0], uV[0], buf0, 1723, 6890, 32, 20670);
  spiral_conv_wmma<32, 32, 1, 2><<<cdivu(ST0, 8), B256, 0, stream>>>(
      buf0, sp[0], deW[3], deB[3], buf1, 6890, ST0);
  // de_last: (32 -> 3), no ELU, straight into d_out
  spiral_conv_wmma<32, 3, 0, 2><<<cdivu(ST0, 8), B256, 0, stream>>>(
      buf1, sp[0], de_last_W, de_last_b, outp + OFF_OUT, 6890, ST0);
}